// MaskedInstanceNorm2d_52733608461084
// MI455X (gfx1250) — compile-verified
//
#include <hip/hip_runtime.h>
#include <stdint.h>

// Problem constants (reference: B=8, C=64, H=512, W=512, fp32)
#define BB 8
#define CC 64
#define HW 262144              // 512*512 pixels per (b,c)
#define HW4 (HW / 4)           // 65536 float4s per (b,c) slice
#define WORDS_PER_B (HW / 32)  // 8192 mask words per image
#define EPSV 1e-5f

typedef __attribute__((ext_vector_type(4))) float f4;
typedef __attribute__((ext_vector_type(4))) unsigned int u4;
typedef __attribute__((ext_vector_type(2))) float v2f;
typedef __attribute__((ext_vector_type(8))) float v8f;

// ---------------------------------------------------------------------------
// Wave32 all-reduce(sum) via V_WMMA_F32_16X16X4_F32 (codegen-confirmed R1).
// a = {v, 0}, B = ones(4x16)  =>  D[m][n] = v_m + v_{m+16}.
// Lanes 0-15 hold D[0..7][n] in the 8 C/D VGPRs, lanes 16-31 hold D[8..15][n],
// so (sum of 8 D VGPRs) + shfl_xor(.,16) == full 32-lane sum on every lane.
// Callers are fully convergent (EXEC all ones).
// ---------------------------------------------------------------------------
__device__ __forceinline__ float wave_allreduce_sum(float v) {
  v2f a;  a[0] = v;     a[1] = 0.0f;
  v2f bm; bm[0] = 1.0f; bm[1] = 1.0f;
  v8f c = {};
  v8f d = __builtin_amdgcn_wmma_f32_16x16x4_f32(
      false, a, false, bm, (short)0, c, false, false);
  float s = d[0] + d[1] + d[2] + d[3] + d[4] + d[5] + d[6] + d[7];
  s += __shfl_xor(s, 16);
  return s;
}

// ---------------------------------------------------------------------------
// Mask bit layout (self-defined; only popcount + k_norm consume it):
//   group g = (f4-index within image-slice) >> 5   (128 pixels per group)
//   word  w = g*4 + j   (j = component 0..3 of the float4)
//   bit   l = f4-index & 31  (lane)
// This lets K1 store raw __ballot words (no cross-lane data movement) and
// lets K3 read one broadcast b128 per wave-iteration.
// ---------------------------------------------------------------------------

// ---------------------------------------------------------------------------
// Kernel 1: one block per (b,c) slice (512 blocks x 256 threads).
// Streams 1 MB with coalesced non-temporal b128 loads; accumulates sum(x),
// sum(x^2) (exact because x==0 wherever mask==0); ORs ballot words into the
// shared per-image mask bitmap.
// ---------------------------------------------------------------------------
__global__ void __launch_bounds__(256)
k_reduce(const f4* __restrict__ x, uint32_t* __restrict__ maskbits,
         float* __restrict__ s1g, float* __restrict__ s2g) {
  const unsigned bc = blockIdx.x;        // 0..511
  const unsigned b = bc >> 6;
  const unsigned t = threadIdx.x;        // 0..255
  const unsigned lane = t & 31;
  const unsigned wave = t >> 5;

  const f4* px = x + (size_t)bc * HW4;
  uint32_t* mb = maskbits + (size_t)b * WORDS_PER_B;

  float ls1 = 0.0f, ls2 = 0.0f;

#pragma unroll 4
  for (unsigned i = 0; i < 256; ++i) {
    f4 v = __builtin_nontemporal_load(px + (size_t)i * 256 + t);
    ls1 += (v.x + v.y) + (v.z + v.w);
    ls2 = fmaf(v.x, v.x, ls2);
    ls2 = fmaf(v.y, v.y, ls2);
    ls2 = fmaf(v.z, v.z, ls2);
    ls2 = fmaf(v.w, v.w, ls2);

    // Wave-uniform ballots: bit l <-> component j of f4 (i*256 + wave*32 + l)
    unsigned b0 = (unsigned)__ballot(v.x != 0.0f);
    unsigned b1 = (unsigned)__ballot(v.y != 0.0f);
    unsigned b2 = (unsigned)__ballot(v.z != 0.0f);
    unsigned b3 = (unsigned)__ballot(v.w != 0.0f);
    unsigned mybal = (lane == 0u) ? b0
                   : (lane == 1u) ? b1
                   : (lane == 2u) ? b2 : b3;
    const unsigned g = i * 8u + wave;
    if (lane < 4u && mybal != 0u) {
      atomicOr(&mb[g * 4u + lane], mybal);
    }
  }

  // Deterministic block reduction: WMMA wave all-reduce + fixed-order combine.
  float w1 = wave_allreduce_sum(ls1);
  float w2 = wave_allreduce_sum(ls2);

  __shared__ float r1[8], r2[8];
  if (lane == 0) { r1[wave] = w1; r2[wave] = w2; }
  __syncthreads();
  if (t == 0) {
    float a1 = 0.0f, a2 = 0.0f;
#pragma unroll
    for (int w = 0; w < 8; ++w) { a1 += r1[w]; a2 += r2[w]; }
    s1g[bc] = a1;
    s2g[bc] = a2;
  }
}

// ---------------------------------------------------------------------------
// Kernel 2: one block per image b (8 blocks x 256 threads).
// popcount mask words -> cnt; fold mean/inv_std/affine/cnt-gate into the
// per-(b,c) valid-pixel scale/shift pair consumed by K3.
// ---------------------------------------------------------------------------
__global__ void __launch_bounds__(256)
k_stats(const uint32_t* __restrict__ maskbits,
        const float* __restrict__ s1g, const float* __restrict__ s2g,
        const float* __restrict__ wgt, const float* __restrict__ bias,
        float* __restrict__ sOn, float* __restrict__ tOn) {
  const unsigned b = blockIdx.x;
  const unsigned t = threadIdx.x;
  const unsigned lane = t & 31, wave = t >> 5;

  const uint32_t* mb = maskbits + (size_t)b * WORDS_PER_B;
  unsigned pc = 0;
#pragma unroll 8
  for (unsigned j = t; j < WORDS_PER_B; j += 256) pc += __popc(mb[j]);

  float w = wave_allreduce_sum((float)pc);  // counts <= 2^18, exact in f32
  __shared__ float rp[8];
  __shared__ float scnt;
  if (lane == 0) rp[wave] = w;
  __syncthreads();
  if (t == 0) {
    float a = 0.0f;
#pragma unroll
    for (int i = 0; i < 8; ++i) a += rp[i];
    scnt = a;
  }
  __syncthreads();

  if (t < CC) {
    const unsigned bc = b * CC + t;
    const float cnt = scnt;
    const float inv = 1.0f / fmaxf(cnt, 1.0f);
    const float mean = s1g[bc] * inv;
    float var = s2g[bc] * inv - mean * mean;
    var = fmaxf(var, 0.0f);
    const float istd = rsqrtf(var + EPSV);
    const float wc = wgt[t];
    const float bb = bias[t];
    const bool on = (cnt > 1.0f);
    // valid-pixel affine:  on -> istd*w * x + (b - mean*istd*w)
    //                      off-> pass-through affine (w, b)
    const float sv = on ? istd * wc : wc;
    const float tv = on ? fmaf(-mean * istd, wc, bb) : bb;
    sOn[bc] = sv;
    tOn[bc] = tv;
  }
}

// ---------------------------------------------------------------------------
// Kernel 3: streaming normalize + affine. One float4 per thread; NT in/out.
// Mask words for the whole wave are one broadcast b128 (g uniform per wave).
// invalid pixels use (w[c], b[c]); valid pixels use the folded (sOn, tOn).
// ---------------------------------------------------------------------------
__global__ void __launch_bounds__(256)
k_norm(const f4* __restrict__ x, const u4* __restrict__ maskbits4,
       const float* __restrict__ sOn, const float* __restrict__ tOn,
       const float* __restrict__ wgt, const float* __restrict__ bias,
       f4* __restrict__ out) {
  const unsigned f = blockIdx.x * 256u + threadIdx.x;  // global float4 index
  const unsigned bc = f >> 16;                         // / HW4
  const unsigned b = bc >> 6;
  const unsigned c = bc & 63u;
  const unsigned fs = f & (HW4 - 1u);                  // f4 index within slice
  const unsigned g = fs >> 5;                          // 128-pixel group
  const unsigned l = fs & 31u;                         // bit index

  f4 v = __builtin_nontemporal_load(x + f);
  const u4 mw = maskbits4[(size_t)b * (WORDS_PER_B / 4) + g];  // wave-broadcast

  const float sv = sOn[bc], tv = tOn[bc];
  const float so = wgt[c],  to = bias[c];

  f4 o;
  o.x = ((mw.x >> l) & 1u) ? fmaf(v.x, sv, tv) : fmaf(v.x, so, to);
  o.y = ((mw.y >> l) & 1u) ? fmaf(v.y, sv, tv) : fmaf(v.y, so, to);
  o.z = ((mw.z >> l) & 1u) ? fmaf(v.z, sv, tv) : fmaf(v.z, so, to);
  o.w = ((mw.w >> l) & 1u) ? fmaf(v.w, sv, tv) : fmaf(v.w, so, to);
  __builtin_nontemporal_store(o, out + f);
}

// ---------------------------------------------------------------------------
extern "C" void kernel_launch(void* const* d_in, const int* in_sizes, int n_in,
                              void* d_out, int out_size, void* d_ws, size_t ws_size,
                              hipStream_t stream) {
  (void)in_sizes; (void)n_in; (void)out_size; (void)ws_size;
  const float* x   = (const float*)d_in[0];
  const float* wgt = (const float*)d_in[1];
  const float* bia = (const float*)d_in[2];
  float* out = (float*)d_out;

  // Workspace layout
  char* ws = (char*)d_ws;
  uint32_t* maskbits = (uint32_t*)ws;                        // 65536 u32 = 256 KB
  float* s1g = (float*)(ws + (size_t)BB * WORDS_PER_B * 4);  // 512 f
  float* s2g = s1g + BB * CC;                                // 512 f
  float* sOn = s2g + BB * CC;                                // 512 f
  float* tOn = sOn + BB * CC;                                // 512 f

  // Mask bits accumulate via atomicOr -> must start at zero each call.
  hipMemsetAsync(maskbits, 0, (size_t)BB * WORDS_PER_B * sizeof(uint32_t), stream);

  k_reduce<<<BB * CC, 256, 0, stream>>>((const f4*)x, maskbits, s1g, s2g);
  k_stats<<<BB, 256, 0, stream>>>(maskbits, s1g, s2g, wgt, bia, sOn, tOn);

  const unsigned n4 = (unsigned)((size_t)BB * CC * HW / 4);  // 33,554,432
  k_norm<<<n4 / 256, 256, 0, stream>>>((const f4*)x, (const u4*)maskbits,
                                       sOn, tOn, wgt, bia, (f4*)out);
}